// StemInference_85478439125794
// MI455X (gfx1250) — compile-verified
//
#include <hip/hip_runtime.h>

// Problem constants (from reference)
#define Bn 16
#define Hn 768
#define Wn 1024
#define Nn (Bn*Hn*Wn)          // 12,582,912
#define Rr 15.0f
#define TVv 0.01f
#define TPp 0.1f

typedef __attribute__((ext_vector_type(2))) float v2f;
typedef __attribute__((ext_vector_type(8))) float v8f;

// ---------------------------------------------------------------- zero scratch
__global__ __launch_bounds__(256) void si_zero(float4* __restrict__ p, int n4) {
  int i = blockIdx.x * 256 + threadIdx.x;
  if (i < n4) p[i] = make_float4(0.f, 0.f, 0.f, 0.f);
}

// ---------------------------------------------------------------- vote scatter
__global__ __launch_bounds__(256) void si_scatter(const float* __restrict__ kp,
                                                  const float* __restrict__ off,
                                                  float* __restrict__ votes) {
  int idx = blockIdx.x * 256 + threadIdx.x;          // < Nn, grid exact
  int x  = idx & (Wn - 1);
  int yb = idx >> 10;                                 // idx / Wn
  int y  = yb % Hn;
  int b  = yb / Hn;
  size_t hw = (size_t)Hn * Wn;
  float w  = kp[idx];
  float o0 = off[(size_t)b * 2 * hw + (size_t)y * Wn + x];
  float o1 = off[(size_t)b * 2 * hw + hw + (size_t)y * Wn + x];
  int vx = __float2int_rn((float)x + Rr * o0);        // v_cvt_i32_f32 RNE == jnp.round
  int vy = __float2int_rn((float)y + Rr * o1);
  if (vx >= 0 && vx < Wn && vy >= 0 && vy < Hn && w >= TVv) {
    const float scale = 1.0f / (3.14159265358979323846f * Rr * Rr);
    // hardware global_atomic_add_f32, no return, agent scope; +-15px => L2-local
    __hip_atomic_fetch_add(&votes[((size_t)b * Hn + vy) * Wn + vx], w * scale,
                           __ATOMIC_RELAXED, __HIP_MEMORY_SCOPE_AGENT);
  }
}

// ------------------------------------------- horizontal 5-sum via WMMA f32 16x16x4
// Block: 16 rows x 128 output cols (8 waves, one 16x16 tile per wave).
// Stage 16x132 input (+-2 col halo) into LDS via CDNA5 async-to-LDS, then
// D(16x16) = A(16x20) x Band(20x16) as 5 chunks of V_WMMA_F32_16X16X4_F32.
#define HS_STRIDE 137   // 137 mod 64 == 9 -> column reads across 16 rows conflict-free
__global__ __launch_bounds__(256) void si_hsum5(const float* __restrict__ src,
                                                float* __restrict__ dst) {
  __shared__ float tile[16 * HS_STRIDE];
  int t    = threadIdx.x;
  int col0 = blockIdx.x * 128;
  int row0 = blockIdx.y * 16;
  const float* s = src + (size_t)blockIdx.z * Hn * Wn;
  float*       d = dst + (size_t)blockIdx.z * Hn * Wn;

  // Stage: tile col c <-> global col (col0 - 2 + c), c in [0,132)
  for (int e = t; e < 16 * 132; e += 256) {
    int r = e / 132, c = e - r * 132;
    int gc = col0 - 2 + c;
    if (gc >= 0 && gc < Wn) {
      unsigned ldsa = (unsigned)(uintptr_t)&tile[r * HS_STRIDE + c];
      unsigned goff = (unsigned)(((row0 + r) * Wn + gc) * 4);
      // ASYNCcnt-tracked DMA into LDS, no VGPR round-trip (GVS addressing)
      asm volatile("global_load_async_to_lds_b32 %0, %1, %2"
                   :: "v"(ldsa), "v"(goff), "s"(s) : "memory");
    } else {
      tile[r * HS_STRIDE + c] = 0.f;    // conv 'SAME' zero padding
    }
  }
  asm volatile("s_wait_asynccnt 0x0" ::: "memory");
  __syncthreads();

  int lane  = t & 31;
  int wv    = t >> 5;
  int lc0   = wv * 16;                 // this wave's tile-local column base
  int m     = lane & 15;               // A row / B col / D col
  int khalf = (lane >> 4) << 1;        // 0 or 2
  v8f acc = {};
#pragma unroll
  for (int k = 0; k < 5; ++k) {
    int kg0 = 4 * k + khalf;           // band row index of component .x
    v2f a;                             // A from LDS: branchless, conflict-free
    a.x = tile[m * HS_STRIDE + lc0 + kg0];
    a.y = tile[m * HS_STRIDE + lc0 + kg0 + 1];
    v2f bf;                            // constant 0/1 band: n <= kg <= n+4
    bf.x = (kg0 - 4 <= m && m <= kg0    ) ? 1.f : 0.f;
    bf.y = (kg0 - 3 <= m && m <= kg0 + 1) ? 1.f : 0.f;
    acc = __builtin_amdgcn_wmma_f32_16x16x4_f32(false, a, false, bf, (short)0,
                                                acc, false, false);
  }
  int rb = row0 + ((lane >> 4) << 3);  // D: VGPR v -> M = v (+8 for upper lanes)
#pragma unroll
  for (int v = 0; v < 8; ++v)
    d[(size_t)(rb + v) * Wn + col0 + lc0 + m] = acc[v];
}

// ------------------------------------------- vertical 5-sum via WMMA f32 16x16x4
// Block: 16 output rows x 128 cols. Stage 20x128 (+-2 row halo, zero OOB) in LDS.
// D(16x16) = Band(16x20) x Bmat(20x16).
#define VS_STRIDE 136   // 2 rows ~ 16 banks -> khalf 0/2 lane halves bank-disjoint
__global__ __launch_bounds__(256) void si_vsum5(const float* __restrict__ src,
                                                float* __restrict__ dst) {
  __shared__ float tile[20 * VS_STRIDE];
  int t    = threadIdx.x;
  int col0 = blockIdx.x * 128;
  int row0 = blockIdx.y * 16;
  const float* s = src + (size_t)blockIdx.z * Hn * Wn;
  float*       d = dst + (size_t)blockIdx.z * Hn * Wn;

  // Stage: tile row r <-> global row (row0 - 2 + r), r in [0,20); coalesced rows
  for (int e = t; e < 20 * 128; e += 256) {
    int r = e >> 7, c = e & 127;
    int gr = row0 - 2 + r;
    tile[r * VS_STRIDE + c] =
        (gr >= 0 && gr < Hn) ? s[(size_t)gr * Wn + col0 + c] : 0.f;
  }
  __syncthreads();

  int lane  = t & 31;
  int wv    = t >> 5;
  int lc0   = wv * 16;
  int m     = lane & 15;
  int khalf = (lane >> 4) << 1;
  v8f acc = {};
#pragma unroll
  for (int k = 0; k < 5; ++k) {
    int kg0 = 4 * k + khalf;
    v2f a;                             // constant 0/1 band: m <= kg <= m+4
    a.x = (m <= kg0     && kg0     <= m + 4) ? 1.f : 0.f;
    a.y = (m <= kg0 + 1 && kg0 + 1 <= m + 4) ? 1.f : 0.f;
    v2f bf;                            // B from LDS: rows kg0, kg0+1
    bf.x = tile[kg0       * VS_STRIDE + lc0 + m];
    bf.y = tile[(kg0 + 1) * VS_STRIDE + lc0 + m];
    acc = __builtin_amdgcn_wmma_f32_16x16x4_f32(false, a, false, bf, (short)0,
                                                acc, false, false);
  }
  int rb = row0 + ((lane >> 4) << 3);
#pragma unroll
  for (int v = 0; v < 8; ++v)
    d[(size_t)(rb + v) * Wn + col0 + lc0 + m] = acc[v];
}

// ---------------------------------------------------- horizontal 17-max (LDS tile)
__global__ __launch_bounds__(256) void si_hmax17(const float* __restrict__ votes,
                                                 float* __restrict__ out) {
  __shared__ float lds[256 + 16];
  int t   = threadIdx.x;
  int x0  = blockIdx.x * 256;
  size_t rowOff = ((size_t)blockIdx.z * Hn + blockIdx.y) * Wn;
  const float* s = votes + rowOff;
  int c = x0 - 8 + t;
  lds[t] = (c >= 0 && c < Wn) ? s[c] : -__builtin_inff();
  if (t < 16) {
    int c2 = x0 + 248 + t;
    lds[256 + t] = (c2 >= 0 && c2 < Wn) ? s[c2] : -__builtin_inff();
  }
  __syncthreads();
  float mx = lds[t];
#pragma unroll
  for (int i = 1; i <= 16; ++i) mx = fmaxf(mx, lds[t + i]);
  out[rowOff + x0 + t] = mx;
}

// -------------------------------- vertical 17-max fused with isclose & threshold
__global__ __launch_bounds__(256) void si_vmax17_peaks(const float* __restrict__ hmax,
                                                       const float* __restrict__ votes,
                                                       float* __restrict__ peaks) {
  int idx = blockIdx.x * 256 + threadIdx.x;
  int x  = idx & (Wn - 1);
  int yb = idx >> 10;
  int y  = yb % Hn;
  int b  = yb / Hn;
  const float* s = hmax + (size_t)b * Hn * Wn;
  float mx = -__builtin_inff();
#pragma unroll
  for (int dy = -8; dy <= 8; ++dy) {
    int r = y + dy;
    if (r >= 0 && r < Hn) mx = fmaxf(mx, s[(size_t)r * Wn + x]);  // coalesced rows
  }
  float v = votes[idx];
  bool close = fabsf(v - mx) <= (1e-8f + 1e-5f * fabsf(mx));      // jnp.isclose
  peaks[idx] = (close && (v > TPp)) ? 1.0f : 0.0f;
}

// --------------------------------------------------------------------- launcher
extern "C" void kernel_launch(void* const* d_in, const int* in_sizes, int n_in,
                              void* d_out, int out_size, void* d_ws, size_t ws_size,
                              hipStream_t stream) {
  const float* kp  = (const float*)d_in[0];   // (16,1,768,1024)
  const float* off = (const float*)d_in[1];   // (16,2,768,1024)
  float* votes = (float*)d_out;               // output 0: conv'd votes (N floats)
  float* out2  = (float*)d_out + (size_t)Nn;  // temp, finally output 1: peaks
  float* ws0   = (float*)d_ws;                // N-float scratch (L2-resident, 50MB)

  si_zero<<<Nn / 4 / 256, 256, 0, stream>>>((float4*)ws0, Nn / 4);
  si_scatter<<<Nn / 256, 256, 0, stream>>>(kp, off, ws0);

  dim3 gs(Wn / 128, Hn / 16, Bn);             // 8 waves/block, 1 tile/wave
  si_hsum5<<<gs, 256, 0, stream>>>(ws0, out2);
  si_vsum5<<<gs, 256, 0, stream>>>(out2, votes);

  si_hmax17<<<dim3(Wn / 256, Hn, Bn), 256, 0, stream>>>(votes, ws0);
  si_vmax17_peaks<<<Nn / 256, 256, 0, stream>>>(ws0, votes, out2);
}